// EnetGnn_69810398429287
// MI455X (gfx1250) — compile-verified
//
#include <hip/hip_runtime.h>
#include <math.h>

typedef __attribute__((ext_vector_type(2))) float v2f;
typedef __attribute__((ext_vector_type(8))) float v8f;

#define HWN 1024   // H*W
#define CN  1024   // channels
#define C4  256    // C/4
#define NB  4      // batch
#define KNN 16     // k

static __device__ __forceinline__ v8f wmma4(v2f a, v2f b, v8f c) {
  // D = A(16x4 f32) x B(4x16 f32) + C(16x16 f32)
  return __builtin_amdgcn_wmma_f32_16x16x4_f32(false, a, false, b, (short)0, c,
                                               false, false);
}

// -------------------------------------------------------------------------
// Kernel 1: r[n] = X_n^T X_n   (X_n = rgb_in[n] as C x HW row-major)
// Wave computes 32(M) x 64(N): 2 A-tiles x 4 B-tiles, 8 accumulators.
// 2048 waves -> 256 blocks x 256 thr.
// -------------------------------------------------------------------------
__global__ void k_gemm_xtx(const float* __restrict__ rgb_in,
                           float* __restrict__ r) {
  const int lane = threadIdx.x & 31;
  const int lm   = lane & 15;
  const int hl   = lane >> 4;                       // lane half (0/1)
  const int gw   = blockIdx.x * 8 + (threadIdx.x >> 5);
  const int n    = gw >> 9;                         // 512 waves per batch
  const int rem  = gw & 511;
  const int i0   = (rem >> 4) * 32;                 // 32-row M tile
  const int j0   = (rem & 15) * 64;                 // 64-col N strip
  const float* X = rgb_in + (size_t)n * CN * HWN;

  v8f accA[4] = {{}, {}, {}, {}};
  v8f accB[4] = {{}, {}, {}, {}};
  for (int c0 = 0; c0 < CN; c0 += 4) {
    const float* Xc = X + (size_t)(c0 + 2 * hl) * HWN;
    v2f a0; a0.x = Xc[i0 + lm];      a0.y = Xc[HWN + i0 + lm];
    v2f a1; a1.x = Xc[i0 + 16 + lm]; a1.y = Xc[HWN + i0 + 16 + lm];
#pragma unroll
    for (int t = 0; t < 4; ++t) {
      v2f b; b.x = Xc[j0 + t * 16 + lm]; b.y = Xc[HWN + j0 + t * 16 + lm];
      accA[t] = wmma4(a0, b, accA[t]);
      accB[t] = wmma4(a1, b, accB[t]);
    }
  }
  float* R = r + ((size_t)n * HWN + i0 + hl * 8) * HWN + j0 + lm;
#pragma unroll
  for (int t = 0; t < 4; ++t)
#pragma unroll
    for (int v = 0; v < 8; ++v) {
      R[(size_t)v * HWN + t * 16]          = accA[t][v];
      R[(size_t)(v + 16) * HWN + t * 16]   = accB[t][v];
    }
}

// -------------------------------------------------------------------------
// Kernel 2: per row of r, indices of the 16 smallest values (ascending,
// ties -> lower index), matching jax.lax.top_k(-r, 16). One wave per row.
// -------------------------------------------------------------------------
__global__ void k_topk16(const float* __restrict__ r, int* __restrict__ idx) {
  const int row  = blockIdx.x * 8 + (threadIdx.x >> 5);  // 0..4095
  const int lane = threadIdx.x & 31;
  const float* p = r + (size_t)row * HWN;

  float v[32];
#pragma unroll
  for (int m = 0; m < 32; ++m) v[m] = p[lane + 32 * m];

  unsigned used = 0u;
  int* orow = idx + row * KNN;
  for (int t = 0; t < KNN; ++t) {
    float best = INFINITY;
    int   bidx = 0x7fffffff;
#pragma unroll
    for (int m = 0; m < 32; ++m) {
      if (!((used >> m) & 1u)) {
        const float val = v[m];
        const int   id  = lane + 32 * m;
        if (val < best || (val == best && id < bidx)) { best = val; bidx = id; }
      }
    }
#pragma unroll
    for (int off = 16; off > 0; off >>= 1) {
      const float ov = __shfl_xor(best, off, 32);
      const int   oi = __shfl_xor(bidx, off, 32);
      if (ov < best || (ov == best && oi < bidx)) { best = ov; bidx = oi; }
    }
    if (lane == 0) orow[t] = bidx;
    if ((bidx & 31) == lane) used |= 1u << (bidx >> 5);
  }
}

// -------------------------------------------------------------------------
// Kernel 3: hid = relu(H0 @ w1 + b1), H0[i][c] = rgb_in[0][c][i].
// M=1024, N=256, K=1024. 256 waves -> 32 blocks.
// -------------------------------------------------------------------------
__global__ void k_mlp_hidden(const float* __restrict__ rgb_in,
                             const float* __restrict__ w1,
                             const float* __restrict__ b1,
                             float* __restrict__ hid) {
  const int lane = threadIdx.x & 31;
  const int lm   = lane & 15;
  const int hl   = lane >> 4;
  const int gw   = blockIdx.x * 8 + (threadIdx.x >> 5);  // 0..255
  const int i0   = (gw >> 2) * 16;
  const int n0   = (gw & 3) * 64;

  v8f acc[4] = {{}, {}, {}, {}};
  for (int c0 = 0; c0 < CN; c0 += 4) {
    const float* Xc = rgb_in + (size_t)(c0 + 2 * hl) * HWN;
    v2f a; a.x = Xc[i0 + lm]; a.y = Xc[HWN + i0 + lm];
    const float* Wc = w1 + (size_t)(c0 + 2 * hl) * C4 + n0 + lm;
#pragma unroll
    for (int t = 0; t < 4; ++t) {
      v2f b; b.x = Wc[t * 16]; b.y = Wc[C4 + t * 16];
      acc[t] = wmma4(a, b, acc[t]);
    }
  }
#pragma unroll
  for (int t = 0; t < 4; ++t) {
    const int   col  = n0 + t * 16 + lm;
    const float bias = b1[col];
#pragma unroll
    for (int v = 0; v < 8; ++v) {
      const int row = i0 + v + 8 * hl;
      hid[(size_t)row * C4 + col] = fmaxf(acc[t][v] + bias, 0.0f);
    }
  }
}

// -------------------------------------------------------------------------
// Kernel 4: F = relu(hid @ w2 + b2). M=1024, N=1024, K=256. 1024 waves.
// -------------------------------------------------------------------------
__global__ void k_mlp_out(const float* __restrict__ hid,
                          const float* __restrict__ w2,
                          const float* __restrict__ b2,
                          float* __restrict__ F) {
  const int lane = threadIdx.x & 31;
  const int lm   = lane & 15;
  const int hl   = lane >> 4;
  const int gw   = blockIdx.x * 8 + (threadIdx.x >> 5);  // 0..1023
  const int i0   = (gw >> 4) * 16;
  const int j0   = (gw & 15) * 64;

  v8f acc[4] = {{}, {}, {}, {}};
  for (int c0 = 0; c0 < C4; c0 += 4) {
    const v2f a = *(const v2f*)(hid + (size_t)(i0 + lm) * C4 + c0 + 2 * hl);
    const float* Wc = w2 + (size_t)(c0 + 2 * hl) * CN + j0 + lm;
#pragma unroll
    for (int t = 0; t < 4; ++t) {
      v2f b; b.x = Wc[t * 16]; b.y = Wc[CN + t * 16];
      acc[t] = wmma4(a, b, acc[t]);
    }
  }
#pragma unroll
  for (int t = 0; t < 4; ++t) {
    const int   col  = j0 + t * 16 + lm;
    const float bias = b2[col];
#pragma unroll
    for (int v = 0; v < 8; ++v) {
      const int row = i0 + v + 8 * hl;
      F[(size_t)row * CN + col] = fmaxf(acc[t][v] + bias, 0.0f);
    }
  }
}

// -------------------------------------------------------------------------
// Kernel 5: g[row] = (1/16) * sum_t F[idx[row][t]]. One block per row.
// -------------------------------------------------------------------------
__global__ void k_gather_mean(const float* __restrict__ F,
                              const int* __restrict__ idx,
                              float* __restrict__ g) {
  const int row = blockIdx.x;       // 0..4095
  const int t   = threadIdx.x;      // 0..255 -> float4 columns
  const int* id = idx + row * KNN;
  float4 acc = make_float4(0.f, 0.f, 0.f, 0.f);
#pragma unroll
  for (int k = 0; k < KNN; ++k) {
    const float4 s = ((const float4*)(F + (size_t)id[k] * CN))[t];
    acc.x += s.x; acc.y += s.y; acc.z += s.z; acc.w += s.w;
  }
  const float sc = 1.0f / 16.0f;
  acc.x *= sc; acc.y *= sc; acc.z *= sc; acc.w *= sc;
  ((float4*)(g + (size_t)row * CN))[t] = acc;
}

// -------------------------------------------------------------------------
// Kernel 6: a[n] = G_n G_n^T. 32x64 per wave; all operands are contiguous
// K-pairs -> b64 vector loads. 2048 waves -> 256 blocks.
// -------------------------------------------------------------------------
__global__ void k_gemm_ggt(const float* __restrict__ g, float* __restrict__ a) {
  const int lane = threadIdx.x & 31;
  const int lm   = lane & 15;
  const int hl   = lane >> 4;
  const int gw   = blockIdx.x * 8 + (threadIdx.x >> 5);
  const int n    = gw >> 9;
  const int rem  = gw & 511;
  const int i0   = (rem >> 4) * 32;
  const int j0   = (rem & 15) * 64;
  const float* G = g + (size_t)n * HWN * CN;

  v8f accA[4] = {{}, {}, {}, {}};
  v8f accB[4] = {{}, {}, {}, {}};
  for (int c0 = 0; c0 < CN; c0 += 4) {
    const size_t ko = c0 + 2 * hl;
    const v2f a0 = *(const v2f*)(G + (size_t)(i0 + lm) * CN + ko);
    const v2f a1 = *(const v2f*)(G + (size_t)(i0 + 16 + lm) * CN + ko);
#pragma unroll
    for (int t = 0; t < 4; ++t) {
      const v2f b = *(const v2f*)(G + (size_t)(j0 + t * 16 + lm) * CN + ko);
      accA[t] = wmma4(a0, b, accA[t]);
      accB[t] = wmma4(a1, b, accB[t]);
    }
  }
  float* A = a + ((size_t)n * HWN + i0 + hl * 8) * HWN + j0 + lm;
#pragma unroll
  for (int t = 0; t < 4; ++t)
#pragma unroll
    for (int v = 0; v < 8; ++v) {
      A[(size_t)v * HWN + t * 16]        = accA[t][v];
      A[(size_t)(v + 16) * HWN + t * 16] = accB[t][v];
    }
}

// -------------------------------------------------------------------------
// Kernel 7: softmax over dim=1 (rows) for each column j of a[n].
// Block = 32 cols x 8 row-groups (coalesced 128B row segments).
// -------------------------------------------------------------------------
__global__ void k_softmax_col(float* __restrict__ a) {
  const int tx = threadIdx.x & 31;   // column in group
  const int ty = threadIdx.x >> 5;   // 0..7
  const int n  = blockIdx.x >> 5;
  const int jg = blockIdx.x & 31;
  float* base  = a + (size_t)n * HWN * HWN + jg * 32 + tx;

  __shared__ float red[8][32];

  float m = -INFINITY;
  for (int i = ty; i < HWN; i += 8) m = fmaxf(m, base[(size_t)i * HWN]);
  red[ty][tx] = m;
  __syncthreads();
  m = red[0][tx];
#pragma unroll
  for (int q = 1; q < 8; ++q) m = fmaxf(m, red[q][tx]);
  __syncthreads();

  float s = 0.f;
  for (int i = ty; i < HWN; i += 8) {
    const float e = __expf(base[(size_t)i * HWN] - m);
    base[(size_t)i * HWN] = e;
    s += e;
  }
  red[ty][tx] = s;
  __syncthreads();
  s = 0.f;
#pragma unroll
  for (int q = 0; q < 8; ++q) s += red[q][tx];
  const float inv = 1.0f / s;
  for (int i = ty; i < HWN; i += 8) base[(size_t)i * HWN] *= inv;
}

// -------------------------------------------------------------------------
// Kernel 8: out[n,c,j] = sum_m Id[n,c,m]*a[n,m,j] + Id[n,c,j].
// 32x64 per wave. A operand is contiguous K-pairs -> b64 loads.
// 2048 waves -> 256 blocks.
// -------------------------------------------------------------------------
__global__ void k_gemm_out(const float* __restrict__ rgb_in,
                           const float* __restrict__ a,
                           float* __restrict__ out) {
  const int lane = threadIdx.x & 31;
  const int lm   = lane & 15;
  const int hl   = lane >> 4;
  const int gw   = blockIdx.x * 8 + (threadIdx.x >> 5);
  const int n    = gw >> 9;
  const int rem  = gw & 511;
  const int c0r  = (rem >> 4) * 32;   // channel-row tile (32 rows)
  const int j0   = (rem & 15) * 64;
  const float* Id = rgb_in + (size_t)n * CN * HWN;
  const float* Am = a + (size_t)n * HWN * HWN;

  v8f accA[4] = {{}, {}, {}, {}};
  v8f accB[4] = {{}, {}, {}, {}};
  for (int m0 = 0; m0 < HWN; m0 += 4) {
    const size_t ko = m0 + 2 * hl;
    const v2f a0 = *(const v2f*)(Id + (size_t)(c0r + lm) * HWN + ko);
    const v2f a1 = *(const v2f*)(Id + (size_t)(c0r + 16 + lm) * HWN + ko);
    const float* Bc = Am + ko * HWN + j0 + lm;
#pragma unroll
    for (int t = 0; t < 4; ++t) {
      v2f b; b.x = Bc[t * 16]; b.y = Bc[HWN + t * 16];
      accA[t] = wmma4(a0, b, accA[t]);
      accB[t] = wmma4(a1, b, accB[t]);
    }
  }
#pragma unroll
  for (int t = 0; t < 4; ++t) {
    const int col = j0 + t * 16 + lm;
#pragma unroll
    for (int v = 0; v < 8; ++v) {
      const int rowA = c0r + v + 8 * hl;
      const int rowB = rowA + 16;
      out[((size_t)n * CN + rowA) * HWN + col] =
          accA[t][v] + Id[(size_t)rowA * HWN + col];
      out[((size_t)n * CN + rowB) * HWN + col] =
          accB[t][v] + Id[(size_t)rowB * HWN + col];
    }
  }
}

// -------------------------------------------------------------------------
extern "C" void kernel_launch(void* const* d_in, const int* in_sizes, int n_in,
                              void* d_out, int out_size, void* d_ws,
                              size_t ws_size, hipStream_t stream) {
  (void)in_sizes; (void)n_in; (void)out_size; (void)ws_size;
  // inputs: 0=cat(unused) 1=rgb_in 2=w1 3=b1 4=w2 5=b2 6=gnn_iterations(=1) 7=k(=16)
  const float* rgb_in = (const float*)d_in[1];
  const float* w1     = (const float*)d_in[2];
  const float* b1     = (const float*)d_in[3];
  const float* w2     = (const float*)d_in[4];
  const float* b2     = (const float*)d_in[5];
  float* out = (float*)d_out;

  float* ws  = (float*)d_ws;
  float* r   = ws;                                   // 4M floats (reused as 'a')
  float* g   = ws + (size_t)4 * 1024 * 1024;         // 4M floats
  float* F   = ws + (size_t)8 * 1024 * 1024;         // 1M floats
  float* hid = ws + (size_t)9 * 1024 * 1024;         // 256K floats
  int*   idx = (int*)(ws + (size_t)9 * 1024 * 1024 + 256 * 1024);  // 64K ints

  k_gemm_xtx   <<<256,  256, 0, stream>>>(rgb_in, r);
  k_topk16     <<<512,  256, 0, stream>>>(r, idx);
  k_mlp_hidden <<<32,   256, 0, stream>>>(rgb_in, w1, b1, hid);
  k_mlp_out    <<<128,  256, 0, stream>>>(hid, w2, b2, F);
  k_gather_mean<<<4096, 256, 0, stream>>>(F, idx, g);
  k_gemm_ggt   <<<256,  256, 0, stream>>>(g, r);      // r reused as 'a'
  k_softmax_col<<<128,  256, 0, stream>>>(r);
  k_gemm_out   <<<256,  256, 0, stream>>>(rgb_in, r, out);
}